// HardKNNMask_5093831213641
// MI455X (gfx1250) — compile-verified
//
#include <hip/hip_runtime.h>
#include <stdint.h>

#define NCOLS 8192
#define BLOCK 256
#define NWAVES (BLOCK / 32)                        // 8 wave32s
#define PER_THREAD (NCOLS / BLOCK)                 // 32 elements per thread
#define CHUNKS ((NCOLS * 4) / (BLOCK * 16))        // 8 x b128 per thread
#define K_RANK 32                                  // 0-based rank of 33rd largest
#define HSTRIDE 257                                // 257 % 64 == 1 -> bank rotation per wave

typedef float  __attribute__((ext_vector_type(4))) floatx4;
typedef unsigned int __attribute__((ext_vector_type(4))) uintx4;

// Order-preserving map f32 bits -> u32 (monotone: bigger float => bigger uint)
__device__ __forceinline__ uint32_t order_f32(uint32_t b) {
    return b ^ ((b & 0x80000000u) ? 0xFFFFFFFFu : 0x80000000u);
}

__global__ __launch_bounds__(BLOCK)
void hard_knn_mask_kernel(const float* __restrict__ sim, float* __restrict__ out) {
    __shared__ uint32_t s_row[NCOLS];               // 32 KB: one full row (raw f32 bits)
    __shared__ uint32_t s_priv[NWAVES * HSTRIDE];   // per-wave histograms, bank-rotated
    __shared__ uint32_t s_hist[256];                // merged histogram
    __shared__ uint32_t s_scan[256];                // suffix-sum workspace
    __shared__ uint32_t s_prefix;                   // selected key prefix (broadcast)
    __shared__ uint32_t s_k;                        // remaining rank (broadcast)

    const uint32_t tid = threadIdx.x;
    const uint32_t wid = tid >> 5;                  // wave32 id
    const uint64_t row = blockIdx.x;
    const uint64_t row_base = (uint64_t)(uintptr_t)sim + row * (uint64_t)(NCOLS * 4);

    // ---------- Stage row: global -> LDS via CDNA5 async data mover ----------
    const uint32_t lds_base = (uint32_t)(uintptr_t)(&s_row[0]);
#pragma unroll
    for (int c = 0; c < CHUNKS; ++c) {
        const uint32_t byte_off = (uint32_t)(c * BLOCK + (int)tid) * 16u;
        const uint64_t gaddr = row_base + byte_off;
        const uint32_t laddr = lds_base + byte_off;
        asm volatile("global_load_async_to_lds_b128 %0, %1, off"
                     :: "v"(laddr), "v"(gaddr)
                     : "memory");
    }
    if (tid == 0) { s_prefix = 0u; s_k = (uint32_t)K_RANK; }
    asm volatile("s_wait_asynccnt 0" ::: "memory");
    __syncthreads();

    // Cache this thread's 32 keys in registers (conflict-free: stride 256 words)
    uint32_t v[PER_THREAD];
#pragma unroll
    for (int i = 0; i < PER_THREAD; ++i)
        v[i] = order_f32(s_row[tid + i * BLOCK]);

    // ---------- 4-pass radix select (8-bit digits, MSB first) ----------
    for (int pass = 0; pass < 4; ++pass) {
        const int shift = 24 - 8 * pass;
        const uint32_t pmask = (pass == 0) ? 0u : (0xFFFFFFFFu << (shift + 8));

        // zero the 8 privatized histograms (8 * 257 = 2056 words, 256 threads)
#pragma unroll
        for (int z = 0; z < (NWAVES * HSTRIDE + BLOCK - 1) / BLOCK; ++z) {
            const uint32_t zi = tid + (uint32_t)(z * BLOCK);
            if (zi < NWAVES * HSTRIDE) s_priv[zi] = 0u;
        }
        __syncthreads();                 // also orders prev-pass s_prefix/s_k writes

        const uint32_t prefix = s_prefix;
        const uint32_t k      = s_k;
        uint32_t* const hpriv = &s_priv[wid * HSTRIDE];

#pragma unroll
        for (int i = 0; i < PER_THREAD; ++i) {
            const uint32_t u = v[i];
            if ((u & pmask) == prefix)
                atomicAdd(&hpriv[(u >> shift) & 0xFFu], 1u);
        }
        __syncthreads();

        // Merge the 8 per-wave histograms: s_hist[b] = sum_w s_priv[w*HSTRIDE + b]
        uint32_t hb = 0u;
#pragma unroll
        for (int w = 0; w < NWAVES; ++w)
            hb += s_priv[w * HSTRIDE + tid];
        s_hist[tid] = hb;
        s_scan[tid] = hb;
        __syncthreads();

        // Parallel inclusive suffix sum: s = sum_{d >= tid} hist[d]
        uint32_t s = hb;
#pragma unroll
        for (int off = 1; off < 256; off <<= 1) {
            const uint32_t add = (tid + (uint32_t)off < 256u) ? s_scan[tid + off] : 0u;
            __syncthreads();
            s += add;
            s_scan[tid] = s;
            __syncthreads();
        }

        // Rank-k digit: unique t with  (s - hist[t]) <= k < s
        const uint32_t h = s_hist[tid];
        const uint32_t E = s - h;        // count of keys with digit > tid
        if (E <= k && k < s) {
            s_prefix = prefix | (tid << shift);
            s_k      = k - E;
        }
    }
    __syncthreads();
    const uint32_t uT = s_prefix;        // exact key of the 33rd-largest element

    // ---------- Masked write-back: LDS -> global, 128-bit NT stores ----------
    const float NEG_INF = -__builtin_inff();
    float* orow = out + row * (uint64_t)NCOLS;
#pragma unroll
    for (int c = 0; c < CHUNKS; ++c) {
        const uint32_t idx = (uint32_t)(c * BLOCK + (int)tid) * 4u;  // word index
        const uintx4 raw = *(const uintx4*)&s_row[idx];
        floatx4 o;
        o.x = (order_f32(raw.x) >= uT) ? __uint_as_float(raw.x) : NEG_INF;
        o.y = (order_f32(raw.y) >= uT) ? __uint_as_float(raw.y) : NEG_INF;
        o.z = (order_f32(raw.z) >= uT) ? __uint_as_float(raw.z) : NEG_INF;
        o.w = (order_f32(raw.w) >= uT) ? __uint_as_float(raw.w) : NEG_INF;
        __builtin_nontemporal_store(o, (floatx4*)&orow[idx]);
    }
}

extern "C" void kernel_launch(void* const* d_in, const int* in_sizes, int n_in,
                              void* d_out, int out_size, void* d_ws, size_t ws_size,
                              hipStream_t stream) {
    (void)n_in; (void)d_ws; (void)ws_size; (void)out_size;
    const float* sim = (const float*)d_in[0];
    float* out = (float*)d_out;
    const int rows = in_sizes[0] / NCOLS;   // 8192
    hard_knn_mask_kernel<<<dim3(rows), dim3(BLOCK), 0, stream>>>(sim, out);
}